// SVD_89850715833158
// MI455X (gfx1250) — compile-verified
//
#include <hip/hip_runtime.h>
#include <math.h>

// ---- problem constants (from reference) ----
#define NPTS   100000
#define NBATCH 64
#define RES    0.2592f
#define CMR    82.8144f   // (640/2 - 0.5) * 0.2592

// ---- tiling ----
#define THREADS        256
#define PTS_PER_THREAD 8
#define CHUNK          (THREADS * PTS_PER_THREAD)          // 2048
#define NCHUNK         ((NPTS + CHUNK - 1) / CHUNK)        // 49
#define LDS_STRIDE     260                                  // avoid 64-bank conflicts (260 % 64 = 4)

typedef __attribute__((ext_vector_type(2))) float v2f;
typedef __attribute__((ext_vector_type(8))) float v8f;

// ---------------------------------------------------------------------------
// Phase 1: per-(batch, chunk) weighted moment accumulation.
// Each thread accumulates 9 moments over 8 points (coalesced float2/float
// loads), partials go to LDS as a 16x256 fp32 matrix P (rows 9..15 zero),
// then wave 0 computes  out(16x1) = P(16x256) * ones(256x1)  with 64 chained
// V_WMMA_F32_16X16X4_F32 (full fp32 -> matches reference precision).
// A-layout per ISA: lanes 0-15 hold K0(V0)/K1(V1) of row M=lane,
//                   lanes 16-31 hold K2(V0)/K3(V1) of row M=lane-16.
// B = all-ones, so its lane layout is irrelevant.
// C/D layout: VGPR r, lanes 0-15 => M=r; lanes 16-31 => M=r+8.
// ---------------------------------------------------------------------------
__global__ __launch_bounds__(THREADS) void svd_moments_kernel(
    const float* __restrict__ kp,   // (128, N, 2) ; src uses rows [::2]
    const float* __restrict__ tg,   // (64,  N, 2)
    const float* __restrict__ wt,   // (64,  1, N)
    float* __restrict__ part)       // (64, NCHUNK, 9)
{
  const int b     = blockIdx.y;
  const int chunk = blockIdx.x;
  const int tid   = threadIdx.x;
  const int base  = chunk * CHUNK;

  const float2* __restrict__ kp2 = (const float2*)kp + (long long)(2 * b) * NPTS;
  const float2* __restrict__ tg2 = (const float2*)tg + (long long)b * NPTS;
  const float*  __restrict__ w   = wt + (long long)b * NPTS;

  float a0 = 0.f, a1 = 0.f, a2 = 0.f,
        a3 = 0.f, a4 = 0.f, a5 = 0.f,
        a6 = 0.f, a7 = 0.f, a8 = 0.f;

#pragma unroll
  for (int i = 0; i < PTS_PER_THREAD; ++i) {
    const int n = base + tid + i * THREADS;
    if (n < NPTS) {
      const float2 k = kp2[n];
      const float2 t = tg2[n];
      const float  ww = w[n];
      // radar-frame conversion: x' = CMR - RES*p1 ; y' = RES*p0 - CMR
      const float sx = CMR - RES * k.y;
      const float sy = RES * k.x - CMR;
      const float tx = CMR - RES * t.y;
      const float ty = RES * t.x - CMR;
      const float wtx = ww * tx;
      const float wty = ww * ty;
      a0 = fmaf(wtx, sx, a0);  a1 = fmaf(wtx, sy, a1);  a2 += wtx;
      a3 = fmaf(wty, sx, a3);  a4 = fmaf(wty, sy, a4);  a5 += wty;
      a6 = fmaf(ww,  sx, a6);  a7 = fmaf(ww,  sy, a7);  a8 += ww;
    }
  }

  __shared__ float P[16][LDS_STRIDE];
  P[0][tid] = a0;  P[1][tid] = a1;  P[2][tid] = a2;
  P[3][tid] = a3;  P[4][tid] = a4;  P[5][tid] = a5;
  P[6][tid] = a6;  P[7][tid] = a7;  P[8][tid] = a8;
#pragma unroll
  for (int f = 9; f < 16; ++f) P[f][tid] = 0.f;
  __syncthreads();

  float* __restrict__ out = part + ((long long)b * NCHUNK + chunk) * 9;

#if __has_builtin(__builtin_amdgcn_wmma_f32_16x16x4_f32)
  if (tid < 32) {
    const int row = tid & 15;
    const int hi  = tid >> 4;             // lanes 16-31 carry K=2,3
    v8f c = {};
    v2f ones; ones[0] = 1.0f; ones[1] = 1.0f;
#pragma unroll 8
    for (int i = 0; i < 64; ++i) {
      const int col = 4 * i + 2 * hi;
      v2f a;
      a[0] = P[row][col];
      a[1] = P[row][col + 1];
      // (neg_a, A, neg_b, B, c_mod, C, reuse_a, reuse_b)
      c = __builtin_amdgcn_wmma_f32_16x16x4_f32(false, a, false, ones,
                                                (short)0, c, false, false);
    }
    // D[m][0]: lane 0 has rows 0..7 in c[0..7]; lane 16 has row 8 in c[0].
    if (tid == 0) {
      out[0] = c[0]; out[1] = c[1]; out[2] = c[2]; out[3] = c[3];
      out[4] = c[4]; out[5] = c[5]; out[6] = c[6]; out[7] = c[7];
    } else if (tid == 16) {
      out[8] = c[0];
    }
  }
#else
  // Fallback: deterministic serial sum per feature.
  if (tid < 9) {
    float s = 0.f;
    for (int t = 0; t < THREADS; ++t) s += P[tid][t];
    out[tid] = s;
  }
#endif
}

// ---------------------------------------------------------------------------
// Phase 2: per batch, fold NCHUNK partials (fixed order -> deterministic),
// then closed-form polar factor of the 2x2 weighted cross-covariance.
// ---------------------------------------------------------------------------
__global__ __launch_bounds__(32) void svd_solve_kernel(
    const float* __restrict__ part,  // (64, NCHUNK, 9)
    float* __restrict__ out)         // [64*9 R^T][64*3 t]
{
  const int b = blockIdx.x;
  const int f = threadIdx.x;
  __shared__ float M[9];

  if (f < 9) {
    float s = 0.f;
    const float* p = part + (long long)b * NCHUNK * 9 + f;
    for (int c = 0; c < NCHUNK; ++c) s += p[c * 9];
    M[f] = s;
  }
  __syncthreads();

  if (f == 0) {
    const float S    = M[8];
    const float invS = 1.0f / S;
    const float sbx = M[6] * invS, sby = M[7] * invS;   // src centroid
    const float tbx = M[2] * invS, tby = M[5] * invS;   // tgt centroid
    // Wc (2x2) = M2x2/S - tbar * sbar^T
    const float a  = M[0] * invS - tbx * sbx;
    const float bb = M[1] * invS - tbx * sby;
    const float cc = M[3] * invS - tby * sbx;
    const float d  = M[4] * invS - tby * sby;
    const float det = a * d - bb * cc;

    float q00, q01, q10, q11, dsign;
    if (det >= 0.f) {
      // closest rotation: maximize cos*(a+d) + sin*(c-b)
      const float hx = a + d, hy = cc - bb;
      const float h  = sqrtf(hx * hx + hy * hy);
      const float co = (h > 0.f) ? hx / h : 1.f;
      const float si = (h > 0.f) ? hy / h : 0.f;
      q00 = co;  q01 = -si;
      q10 = si;  q11 = co;
      dsign = 1.f;
    } else {
      // closest reflection: maximize p*(a-d) + q*(b+c)  (det<0 => h>0)
      const float hx = a - d, hy = bb + cc;
      const float h  = sqrtf(hx * hx + hy * hy);
      const float p  = hx / h, q = hy / h;
      q00 = p;   q01 = q;
      q10 = q;   q11 = -p;
      dsign = -1.f;
    }

    // t1 = sbar - R^T * tbar ;  t_out = -R * t1   (z components stay 0)
    const float t1x = sbx - (q00 * tbx + q10 * tby);
    const float t1y = sby - (q01 * tbx + q11 * tby);
    const float tox = -(q00 * t1x + q01 * t1y);
    const float toy = -(q10 * t1x + q11 * t1y);

    float* Ro = out + b * 9;            // R^T, row-major
    Ro[0] = q00; Ro[1] = q10; Ro[2] = 0.f;
    Ro[3] = q01; Ro[4] = q11; Ro[5] = 0.f;
    Ro[6] = 0.f; Ro[7] = 0.f; Ro[8] = dsign;
    float* To = out + NBATCH * 9 + b * 3;
    To[0] = tox; To[1] = toy; To[2] = 0.f;
  }
}

extern "C" void kernel_launch(void* const* d_in, const int* in_sizes, int n_in,
                              void* d_out, int out_size, void* d_ws, size_t ws_size,
                              hipStream_t stream) {
  const float* kp = (const float*)d_in[0];   // (128, 100000, 2)
  const float* tg = (const float*)d_in[1];   // (64, 100000, 2)
  const float* wt = (const float*)d_in[2];   // (64, 1, 100000)
  float* part = (float*)d_ws;                // 64*49*9 floats = 112,896 B

  dim3 grid(NCHUNK, NBATCH);
  svd_moments_kernel<<<grid, THREADS, 0, stream>>>(kp, tg, wt, part);
  svd_solve_kernel<<<NBATCH, 32, 0, stream>>>(part, (float*)d_out);
}